// LineScorer_60017872995033
// MI455X (gfx1250) — compile-verified
//
#include <hip/hip_runtime.h>
#include <math.h>

// ---------------------------------------------------------------------------
// LineScorer fused kernel for MI455X (gfx1250, wave32, WMMA).
//
// Roofline: ~4 GFLOP vs ~512MB of L2-resident gather traffic (feature map is
// 32MB << 192MB L2) -> gather-bound. Keep everything f32 (matches reference),
// use V_WMMA_F32_16X16X4_F32 for the conv GEMMs, fuse the whole pipeline so
// pooled activations never leave LDS.
// ---------------------------------------------------------------------------

#define B_     4
#define NL_    2048
#define C_     128
#define H_     128
#define W_     128
#define P_     64          // bottleneck channels
#define NPTS0  32
#define NPTS1  8
#define NLINES (B_ * NL_)  // 8192
#define LT     16          // lines per workgroup
#define NCOLS  (LT * NPTS1) // 128 GEMM columns per tile
#define FCIN   (C_ * NPTS1 + 5) // 1029

typedef float v2f __attribute__((ext_vector_type(2)));
typedef float v8f __attribute__((ext_vector_type(8)));

// LDS layout (floats). Pitches chosen so (pitch % 64) == 4 -> column reads
// across 16 rows hit 16 distinct banks (A-fragment loads).
#define PITCH_X   132
#define PITCH_W1  132
#define PITCH_W2  68
#define PITCH_W3  68
#define OFF_SX    0
#define OFF_A     (OFF_SX + C_ * PITCH_X)        // W1 (64x132), reused as Y2
#define OFF_Y1    (OFF_A + P_ * PITCH_W1)        // 64x132
#define OFF_W2    (OFF_Y1 + P_ * PITCH_W1)       // 3 x 64 x 68
#define OFF_W3    (OFF_W2 + 3 * P_ * PITCH_W2)   // 128 x 68
#define OFF_BN    (OFF_W3 + C_ * PITCH_W3)       // s1,t1(128) s2,t2,s3,t3(64)
#define SMEM_FLOATS (OFF_BN + 2 * C_ + 4 * P_)   // 56064 floats = 224256 B

static __device__ __forceinline__ v8f wmma4(v2f a, v2f b, v8f c) {
  // D = A(16x4,f32) * B(4x16,f32) + C(16x16,f32); 8-arg CDNA5 form.
  return __builtin_amdgcn_wmma_f32_16x16x4_f32(
      /*neg_a=*/false, a, /*neg_b=*/false, b,
      /*c_mod=*/(short)0, c, /*reuse_a=*/false, /*reuse_b=*/false);
}

__global__ __launch_bounds__(256, 1) void line_scorer_fused(
    const float* __restrict__ fm,   const float* __restrict__ lines,
    const float* __restrict__ bn1g, const float* __restrict__ bn1b,
    const float* __restrict__ w1,   const float* __restrict__ b1,
    const float* __restrict__ bn2g, const float* __restrict__ bn2b,
    const float* __restrict__ w2,   const float* __restrict__ b2,
    const float* __restrict__ bn3g, const float* __restrict__ bn3b,
    const float* __restrict__ w3,   const float* __restrict__ b3,
    const float* __restrict__ fcw,  const float* __restrict__ fcb,
    float* __restrict__ out)
{
  extern __shared__ float smem[];
  float* sX  = smem + OFF_SX;   // raw pooled x, 128 x (16 lines * 8)
  float* sA  = smem + OFF_A;    // conv1 weights, later Y2
  float* sY1 = smem + OFF_Y1;   // conv1 activations
  float* sW2 = smem + OFF_W2;   // conv2 weights [tap][o][i]
  float* sW3 = smem + OFF_W3;   // conv3 weights
  float* s1 = smem + OFF_BN;  float* t1 = s1 + C_;
  float* s2 = t1 + C_;        float* t2 = s2 + P_;
  float* s3 = t2 + P_;        float* t3 = s3 + P_;

  const int tid  = threadIdx.x;
  const int wave = tid >> 5;          // 8 waves
  const int lane = tid & 31;
  const int lrow = lane & 15;
  const int lhi  = lane >> 4;
  const int lineBase = blockIdx.x * LT;
  const int b = lineBase / NL_;
  const float rs = rsqrtf(1.0f + 1e-5f);

  __builtin_prefetch(fcw, 0, 0);      // global_prefetch_b8: FC weights -> L2
  __builtin_prefetch(w2, 0, 0);

  // ---- phase 0: stage weights + folded BN params into LDS -----------------
  for (int i = tid; i < P_ * C_; i += 256)           // conv1_w (64,128,1)
    sA[(i >> 7) * PITCH_W1 + (i & 127)] = w1[i];
  for (int i = tid; i < P_ * P_ * 3; i += 256) {     // conv2_w (64,64,3)
    int o = i / 192, rem = i - o * 192, ii = rem / 3, tap = rem - ii * 3;
    sW2[tap * (P_ * PITCH_W2) + o * PITCH_W2 + ii] = w2[i];
  }
  for (int i = tid; i < C_ * P_; i += 256)           // conv3_w (128,64,1)
    sW3[(i >> 6) * PITCH_W3 + (i & 63)] = w3[i];
  if (tid < C_) { s1[tid] = bn1g[tid] * rs; t1[tid] = bn1b[tid]; }
  if (tid < P_) { s2[tid] = bn2g[tid] * rs; t2[tid] = bn2b[tid];
                  s3[tid] = bn3g[tid] * rs; t3[tid] = bn3b[tid]; }

  // ---- phase 1: LOI gather + maxpool(4) into sX ---------------------------
  // lane = point index (consecutive points -> nearby addresses -> wave-level
  // sector coalescing out of L2); wave handles 2 lines x 128 channels.
  const float* fmb = fm + (size_t)b * (C_ * H_ * W_);
  for (int rep = 0; rep < 2; ++rep) {
    const int ll = wave + 8 * rep;
    const int gl = lineBase + ll;
    const float xu = lines[gl * 4 + 0], yu = lines[gl * 4 + 1];
    const float xv = lines[gl * 4 + 2], yv = lines[gl * 4 + 3];
    const float lam = (float)lane * (1.0f / 31.0f);
    const float px = xu * lam + xv * (1.0f - lam) - 0.5f;
    const float py = yu * lam + yv * (1.0f - lam) - 0.5f;
    const float px0 = fminf(fmaxf(floorf(px), 0.f), 127.f);
    const float py0 = fminf(fmaxf(floorf(py), 0.f), 127.f);
    const float px1 = fminf(px0 + 1.f, 127.f);
    const float py1 = fminf(py0 + 1.f, 127.f);
    const int ix0 = (int)px0, iy0 = (int)py0, ix1 = (int)px1, iy1 = (int)py1;
    // weight pairing replicates the reference exactly
    const float wa = (px1 - px) * (py1 - py);   // (y0,x0)
    const float wb = (px - px0) * (py1 - py);   // (y1,x0)
    const float wc = (px1 - px) * (py - py0);   // (y0,x1)
    const float wd = (px - px0) * (py - py0);   // (y1,x1)
    const int i00 = iy0 * W_ + ix0, i10 = iy1 * W_ + ix0;
    const int i01 = iy0 * W_ + ix1, i11 = iy1 * W_ + ix1;
    const int j = lane >> 2;
#pragma unroll 4
    for (int c = 0; c < C_; ++c) {
      const float* fc = fmb + c * (H_ * W_);
      float v = fc[i00] * wa + fc[i10] * wb + fc[i01] * wc + fc[i11] * wd;
      v = fmaxf(v, __shfl_xor(v, 1, 32));     // maxpool over 4 consecutive pts
      v = fmaxf(v, __shfl_xor(v, 2, 32));
      if ((lane & 3) == 0) sX[c * PITCH_X + ll * NPTS1 + j] = v;
    }
  }
  __syncthreads();

  const v8f vzero = {0.f, 0.f, 0.f, 0.f, 0.f, 0.f, 0.f, 0.f};

  // ---- conv1: D(64x128) = W1(64x128) @ relu(bn1(X))(128x128) --------------
  {
    const int m = wave & 3, h = wave >> 2, colb = h * 64;
    v8f acc[4];
#pragma unroll
    for (int nt = 0; nt < 4; ++nt) acc[nt] = vzero;
    for (int kk = 0; kk < C_; kk += 4) {
      const int ka = kk + 2 * lhi;
      v2f a = { sA[(m * 16 + lrow) * PITCH_W1 + ka],
                sA[(m * 16 + lrow) * PITCH_W1 + ka + 1] };
      const float sc0 = s1[ka], sh0 = t1[ka], sc1 = s1[ka + 1], sh1 = t1[ka + 1];
#pragma unroll
      for (int nt = 0; nt < 4; ++nt) {
        const int col = colb + nt * 16 + lrow;
        v2f bb = { fmaxf(sX[ka * PITCH_X + col] * sc0 + sh0, 0.f),
                   fmaxf(sX[(ka + 1) * PITCH_X + col] * sc1 + sh1, 0.f) };
        acc[nt] = wmma4(a, bb, acc[nt]);
      }
    }
    float bias[8], sc[8], sh[8];
#pragma unroll
    for (int r = 0; r < 8; ++r) {
      const int M = m * 16 + r + 8 * lhi;
      bias[r] = b1[M]; sc[r] = s2[M]; sh[r] = t2[M];
    }
#pragma unroll
    for (int nt = 0; nt < 4; ++nt) {
      const int col = colb + nt * 16 + lrow;
#pragma unroll
      for (int r = 0; r < 8; ++r) {
        const int M = m * 16 + r + 8 * lhi;
        sY1[M * PITCH_W1 + col] =
            fmaxf((acc[nt][r] + bias[r]) * sc[r] + sh[r], 0.f);
      }
    }
  }
  __syncthreads();

  // ---- conv2 (k=3, pad 1): 3 shifted-B GEMM taps, per-line zero edges -----
  {
    const int m = wave & 3, h = wave >> 2, colb = h * 64;
    v8f acc[4];
#pragma unroll
    for (int nt = 0; nt < 4; ++nt) acc[nt] = vzero;
    for (int tap = 0; tap < 3; ++tap) {
      const int dk = tap - 1;
      const float* wA = sW2 + tap * (P_ * PITCH_W2);
      for (int kk = 0; kk < P_; kk += 4) {
        const int ka = kk + 2 * lhi;
        v2f a = { wA[(m * 16 + lrow) * PITCH_W2 + ka],
                  wA[(m * 16 + lrow) * PITCH_W2 + ka + 1] };
#pragma unroll
        for (int nt = 0; nt < 4; ++nt) {
          const int col = colb + nt * 16 + lrow;
          const bool ok = ((unsigned)((col & 7) + dk) < 8u);
          const int cs = col + dk;
          v2f bb = { ok ? sY1[ka * PITCH_W1 + cs] : 0.f,
                     ok ? sY1[(ka + 1) * PITCH_W1 + cs] : 0.f };
          acc[nt] = wmma4(a, bb, acc[nt]);
        }
      }
    }
    float bias[8], sc[8], sh[8];
#pragma unroll
    for (int r = 0; r < 8; ++r) {
      const int M = m * 16 + r + 8 * lhi;
      bias[r] = b2[M]; sc[r] = s3[M]; sh[r] = t3[M];
    }
#pragma unroll
    for (int nt = 0; nt < 4; ++nt) {
      const int col = colb + nt * 16 + lrow;
#pragma unroll
      for (int r = 0; r < 8; ++r) {
        const int M = m * 16 + r + 8 * lhi;
        sA[M * PITCH_W1 + col] =            // sA reused as Y2 (W1 dead now)
            fmaxf((acc[nt][r] + bias[r]) * sc[r] + sh[r], 0.f);
      }
    }
  }
  __syncthreads();

  // ---- conv3 + residual + relu (for FC): D(128x128) = W3(128x64) @ Y2 -----
  {
    const int m = wave;                      // 8 M-strips of 16
    v8f acc[8];
#pragma unroll
    for (int nt = 0; nt < 8; ++nt) acc[nt] = vzero;
    for (int kk = 0; kk < P_; kk += 4) {
      const int ka = kk + 2 * lhi;
      v2f a = { sW3[(m * 16 + lrow) * PITCH_W3 + ka],
                sW3[(m * 16 + lrow) * PITCH_W3 + ka + 1] };
#pragma unroll
      for (int nt = 0; nt < 8; ++nt) {
        const int col = nt * 16 + lrow;
        v2f bb = { sA[ka * PITCH_W1 + col], sA[(ka + 1) * PITCH_W1 + col] };
        acc[nt] = wmma4(a, bb, acc[nt]);
      }
    }
    float bias[8];
#pragma unroll
    for (int r = 0; r < 8; ++r) bias[r] = b3[m * 16 + r + 8 * lhi];
#pragma unroll
    for (int nt = 0; nt < 8; ++nt) {
      const int col = nt * 16 + lrow;
#pragma unroll
      for (int r = 0; r < 8; ++r) {
        const int M = m * 16 + r + 8 * lhi;
        const float z = fmaxf(sX[M * PITCH_X + col] + acc[nt][r] + bias[r], 0.f);
        sX[M * PITCH_X + col] = z;           // in-place: each elem 1 owner
      }
    }
  }
  __syncthreads();

  // ---- FC (1029 -> 4) + softmax: 16 lanes per line, shfl-reduce -----------
  {
    const int ll = tid >> 4, j = tid & 15;
    const int gl = lineBase + ll;
    float a0 = 0.f, a1 = 0.f, a2 = 0.f, a3 = 0.f;
    for (int e = j; e < C_ * NPTS1; e += 16) {     // coalesced fcw reads
      const int c = e >> 3, pos = e & 7;
      const float z = sX[c * PITCH_X + ll * NPTS1 + pos];
      a0 += z * fcw[0 * FCIN + e];
      a1 += z * fcw[1 * FCIN + e];
      a2 += z * fcw[2 * FCIN + e];
      a3 += z * fcw[3 * FCIN + e];
    }
#pragma unroll
    for (int msk = 8; msk >= 1; msk >>= 1) {
      a0 += __shfl_xor(a0, msk, 16);
      a1 += __shfl_xor(a1, msk, 16);
      a2 += __shfl_xor(a2, msk, 16);
      a3 += __shfl_xor(a3, msk, 16);
    }
    if (j == 0) {
      const float xu = lines[gl * 4 + 0], yu = lines[gl * 4 + 1];
      const float xv = lines[gl * 4 + 2], yv = lines[gl * 4 + 3];
      const float dx = xu - xv, dy = yu - yv;
      const float u2v = fmaxf(sqrtf(dx * dx + dy * dy), 1e-6f);
      const float f[5] = { fmaxf(xu * (1.f / 128.f), 0.f),
                           fmaxf(yu * (1.f / 128.f), 0.f),
                           fmaxf(xv * (1.f / 128.f), 0.f),
                           fmaxf(yv * (1.f / 128.f), 0.f),
                           fmaxf(u2v, 0.f) };
      float lg[4] = { a0, a1, a2, a3 };
#pragma unroll
      for (int r = 0; r < 4; ++r) {
        float s = lg[r] + fcb[r];
#pragma unroll
        for (int q = 0; q < 5; ++q) s += f[q] * fcw[r * FCIN + 1024 + q];
        lg[r] = s;
      }
      const float mx = fmaxf(fmaxf(lg[0], lg[1]), fmaxf(lg[2], lg[3]));
      const float e0 = __expf(lg[0] - mx), e1 = __expf(lg[1] - mx);
      const float e2 = __expf(lg[2] - mx), e3 = __expf(lg[3] - mx);
      const float inv = 1.f / (e0 + e1 + e2 + e3);
      out[gl * 4 + 0] = lg[0]; out[gl * 4 + 1] = lg[1];
      out[gl * 4 + 2] = lg[2]; out[gl * 4 + 3] = lg[3];
      float* prob = out + NLINES * 4;
      prob[gl * 4 + 0] = e0 * inv; prob[gl * 4 + 1] = e1 * inv;
      prob[gl * 4 + 2] = e2 * inv; prob[gl * 4 + 3] = e3 * inv;
    }
  }
}

extern "C" void kernel_launch(void* const* d_in, const int* in_sizes, int n_in,
                              void* d_out, int out_size, void* d_ws, size_t ws_size,
                              hipStream_t stream) {
  (void)in_sizes; (void)n_in; (void)out_size; (void)d_ws; (void)ws_size;
  const float* fm   = (const float*)d_in[0];
  const float* ln   = (const float*)d_in[1];
  const float* bn1g = (const float*)d_in[2];
  const float* bn1b = (const float*)d_in[3];
  const float* w1   = (const float*)d_in[4];
  const float* b1   = (const float*)d_in[5];
  const float* bn2g = (const float*)d_in[6];
  const float* bn2b = (const float*)d_in[7];
  const float* w2   = (const float*)d_in[8];
  const float* b2   = (const float*)d_in[9];
  const float* bn3g = (const float*)d_in[10];
  const float* bn3b = (const float*)d_in[11];
  const float* w3   = (const float*)d_in[12];
  const float* b3   = (const float*)d_in[13];
  const float* fcw  = (const float*)d_in[14];
  const float* fcb  = (const float*)d_in[15];
  float* out = (float*)d_out;

  const size_t shmem = SMEM_FLOATS * sizeof(float); // 224256 B < 320 KB/WGP
  line_scorer_fused<<<NLINES / LT, 256, shmem, stream>>>(
      fm, ln, bn1g, bn1b, w1, b1, bn2g, bn2b, w2, b2,
      bn3g, bn3b, w3, b3, fcw, fcb, out);
}